// Dynamic_sampling_86526411145631
// MI455X (gfx1250) — compile-verified
//
#include <hip/hip_runtime.h>

// Problem constants (match reference)
#define BATCH 8
#define CCH   4
#define PN    8192
#define SN    2048
#define KK    20

typedef __attribute__((ext_vector_type(2))) float v2f;
typedef __attribute__((ext_vector_type(8))) float v8f;

// ---------------------------------------------------------------------------
// Phase 1: replicate perm = argsort(rand)[:, :SN] via stable rank counting.
// Grid = BATCH * (PN/1024) blocks of 1024 threads; each thread owns ONE p,
// scans the whole rand row from LDS (uniform broadcast reads).
// rank(p) = #{j : rand[j] < rand[p]  ||  (rand[j] == rand[p] && j < p)}
// If rank < SN, stage point p's coords (+|s|^2) into selPts[b][rank].
// ---------------------------------------------------------------------------
__global__ __launch_bounds__(1024) void sample_kernel(
    const float* __restrict__ x, const float* __restrict__ rnd,
    float4* __restrict__ selPts) {
  __shared__ float lr[PN];
  const int chunksPerBatch = PN / 1024;  // 8
  const int b = blockIdx.x / chunksPerBatch;
  const int p = (blockIdx.x % chunksPerBatch) * 1024 + threadIdx.x;

  for (int j = threadIdx.x; j < PN; j += 1024) lr[j] = rnd[(size_t)b * PN + j];
  __syncthreads();

  const float rp = lr[p];
  int rank = 0;
#pragma unroll 8
  for (int j = 0; j < PN; ++j) {
    const float rv = lr[j];
    rank += (rv < rp || (rv == rp && j < p)) ? 1 : 0;
  }

  if (rank < SN) {
    const float* xb = x + (size_t)b * CCH * PN;
    const float x0 = xb[p], x1 = xb[PN + p], x2 = xb[2 * PN + p];
    selPts[(size_t)b * SN + rank] =
        make_float4(x0, x1, x2, x0 * x0 + x1 * x1 + x2 * x2);
  }
}

// ---------------------------------------------------------------------------
// Phase 2: fused WMMA distance streaming + per-column register top-20.
// Block = 256 threads (8 wave32). Each wave owns 32 sample columns
// (two 16-col WMMA B tiles) and streams all P rows in 16-row A tiles.
// Score = 2*inner - |q|^2  (dropping the per-column -|s|^2 constant
// preserves the top_k ordering within a column).
// ---------------------------------------------------------------------------
__global__ __launch_bounds__(256) void knn_topk_kernel(
    const float* __restrict__ x, const float4* __restrict__ selPts,
    int* __restrict__ topk) {
  __shared__ float pdT[8][16][33];  // per-wave 16x32 score tile (padded)
  __shared__ float qn[8][16];       // per-wave |q|^2 for the 16 rows

  const int tid = threadIdx.x;
  const int lane = tid & 31;
  const int wave = tid >> 5;
  const int laneLo = lane & 15;
  const bool hi = lane >= 16;

  const int blocksPerBatch = SN / 256;  // 8
  const int b = blockIdx.x / blocksPerBatch;
  const int s0 = (blockIdx.x % blocksPerBatch) * 256 + wave * 32;

  const float* xb = x + (size_t)b * CCH * PN;

  // B-matrix tiles (4x16 K-major): lane<16 -> {B[0][n],B[1][n]} = (x,y),
  //                                lane>=16 -> {B[2][n],B[3][n]} = (z,0)
  const float4 sp0 = selPts[(size_t)b * SN + s0 + laneLo];
  const float4 sp1 = selPts[(size_t)b * SN + s0 + 16 + laneLo];
  v2f bm0, bm1;
  bm0[0] = hi ? sp0.z : sp0.x;
  bm0[1] = hi ? 0.0f : sp0.y;
  bm1[0] = hi ? sp1.z : sp1.x;
  bm1[1] = hi ? 0.0f : sp1.y;

  // Register-resident sorted top-20 (descending). Stable insertion matches
  // jax.lax.top_k tie-breaking (lowest index wins among equal values).
  float tv[KK];
  int ti[KK];
#pragma unroll
  for (int j = 0; j < KK; ++j) {
    tv[j] = -__builtin_inff();
    ti[j] = 0;
  }

  for (int p0 = 0; p0 < PN; p0 += 16) {
    const int pr = p0 + laneLo;

    // A-matrix tile (16x4): lane<16 -> {x0,x1}, lane>=16 -> {x2,0}
    const float x2 = xb[2 * PN + pr];
    float x0 = 0.0f, x1 = 0.0f;
    if (!hi) {
      x0 = xb[pr];
      x1 = xb[PN + pr];
      qn[wave][laneLo] = x0 * x0 + x1 * x1 + x2 * x2;
    }
    v2f am;
    am[0] = hi ? x2 : x0;
    am[1] = hi ? 0.0f : x1;

    v8f acc0 = {};
    v8f acc1 = {};
    acc0 = __builtin_amdgcn_wmma_f32_16x16x4_f32(false, am, false, bm0,
                                                 (short)0, acc0, false, false);
    acc1 = __builtin_amdgcn_wmma_f32_16x16x4_f32(false, am, false, bm1,
                                                 (short)0, acc1, false, false);

    // D layout: VGPR r = row r (lanes 0-15) / row r+8 (lanes 16-31), col = lane&15
#pragma unroll
    for (int r = 0; r < 8; ++r) {
      const int row = r + (hi ? 8 : 0);
      pdT[wave][row][laneLo] = acc0[r];
      pdT[wave][row][laneLo + 16] = acc1[r];
    }
    __syncthreads();

    // Each lane owns column (s0 + lane); scan the 16 new rows.
#pragma unroll
    for (int r = 0; r < 16; ++r) {
      const float v = 2.0f * pdT[wave][r][lane] - qn[wave][r];
      const int pidx = p0 + r;
      // Rare-path guard; body is a BRANCHLESS shift-insert (pure v_cndmask,
      // no whole-array copies).
      if (v > tv[KK - 1]) {
        bool c[KK];
#pragma unroll
        for (int j = 0; j < KK; ++j) c[j] = v > tv[j];
#pragma unroll
        for (int j = KK - 1; j > 0; --j) {
          tv[j] = c[j - 1] ? tv[j - 1] : (c[j] ? v : tv[j]);
          ti[j] = c[j - 1] ? ti[j - 1] : (c[j] ? pidx : ti[j]);
        }
        tv[0] = c[0] ? v : tv[0];
        ti[0] = c[0] ? pidx : ti[0];
      }
    }
    __syncthreads();
  }

  const int s = s0 + lane;
#pragma unroll
  for (int j = 0; j < KK; ++j)
    topk[((size_t)b * SN + s) * KK + j] = ti[j];
}

// ---------------------------------------------------------------------------
// Phase 3: gather with the reference's (B,K,S,C)->reshape(B,S,K,C) quirk:
// out[b,c,s,k] = x[b,c, topk[b][f % S][f / S]] with f = s*K + k.
// ---------------------------------------------------------------------------
__global__ void gather_kernel(const float* __restrict__ x,
                              const int* __restrict__ topk,
                              float* __restrict__ out) {
  const int total = BATCH * CCH * SN * KK;
  const int o = blockIdx.x * blockDim.x + threadIdx.x;
  if (o >= total) return;
  const int kk_new = o % KK;
  const int s_new = (o / KK) % SN;
  const int c = (o / (KK * SN)) % CCH;
  const int b = o / (KK * SN * CCH);
  const int f = s_new * KK + kk_new;
  const int k_old = f / SN;
  const int s_old = f % SN;
  const int p = topk[((size_t)b * SN + s_old) * KK + k_old];
  out[o] = x[((size_t)b * CCH + c) * PN + p];
}

extern "C" void kernel_launch(void* const* d_in, const int* in_sizes, int n_in,
                              void* d_out, int out_size, void* d_ws,
                              size_t ws_size, hipStream_t stream) {
  const float* x = (const float*)d_in[0];
  const float* rnd = (const float*)d_in[1];
  // s_num / k (d_in[2], d_in[3]) are compile-time constants SN, KK.

  float4* selPts = (float4*)d_ws;  // B*S float4 = 256 KB
  int* topk = (int*)((char*)d_ws + (size_t)BATCH * SN * sizeof(float4));
  float* out = (float*)d_out;

  sample_kernel<<<BATCH * (PN / 1024), 1024, 0, stream>>>(x, rnd, selPts);
  knn_topk_kernel<<<BATCH * (SN / 256), 256, 0, stream>>>(x, selPts, topk);
  const int total = BATCH * CCH * SN * KK;
  gather_kernel<<<(total + 255) / 256, 256, 0, stream>>>(x, topk, out);
}